// StandaloneGatedGCNLayer_56014963475030
// MI455X (gfx1250) — compile-verified
//
#include <hip/hip_runtime.h>
#include <cstdint>

#define EPS 1e-5f

typedef __attribute__((ext_vector_type(2))) float v2f;
typedef __attribute__((ext_vector_type(8))) float v8f;

// ---------------------------------------------------------------------------
// WMMA helper: D = A(16x4 f32) * B(4x16 f32) + C(16x16 f32)
// ---------------------------------------------------------------------------
__device__ __forceinline__ v8f wmma4(v2f a, v2f b, v8f c) {
  return __builtin_amdgcn_wmma_f32_16x16x4_f32(
      /*neg_a=*/false, a, /*neg_b=*/false, b,
      /*c_mod=*/(short)0, c, /*reuse_a=*/false, /*reuse_b=*/false);
}

// A fragment: rows [row0,row0+16), K window [k0,k0+4), row-major, ld columns.
// ISA layout: lanes 0-15 hold M=lane, VGPR0=K{0 or 2}, VGPR1=K{1 or 3}.
__device__ __forceinline__ v2f load_a(const float* M, int row0, int k0,
                                      int lane, int maxRow, int ld) {
  int m  = lane & 15;
  int kk = (lane >> 4) << 1;   // 0 or 2
  int r  = row0 + m; if (r > maxRow - 1) r = maxRow - 1;
  const float* p = M + (size_t)r * ld + k0 + kk;
  v2f v; v.x = p[0]; v.y = p[1]; return v;
}

// B fragment for out = in @ W^T  (B[k][n] = W[n][k]); W row-major [outdim][ld].
__device__ __forceinline__ v2f load_bt(const float* W, int n0, int k0,
                                       int lane, int ld) {
  int n  = lane & 15;
  int kk = (lane >> 4) << 1;
  const float* p = W + (size_t)(n0 + n) * ld + k0 + kk;
  v2f v; v.x = p[0]; v.y = p[1]; return v;
}

// C/D layout: VGPR r -> (M = r + 8*(lane>=16), N = n0 + (lane&15))
__device__ __forceinline__ void store_d_full(float* out, int row0, int n0,
                                             int lane, const v8f& d, int ld) {
  int half = lane >> 4;
  int n    = lane & 15;
#pragma unroll
  for (int r = 0; r < 8; ++r)
    out[(size_t)(row0 + r + half * 8) * ld + n0 + n] = d[r];
}

__device__ __forceinline__ void store_d_guard(float* out, int row0, int n0,
                                              int lane, const v8f& d,
                                              int maxRow, int ld) {
  int half = lane >> 4;
  int n    = lane & 15;
#pragma unroll
  for (int r = 0; r < 8; ++r) {
    int row = row0 + r + half * 8;
    if (row < maxRow) out[(size_t)row * ld + n0 + n] = d[r];
  }
}

// ---------------------------------------------------------------------------
// K0: zero aggr + stats
// ---------------------------------------------------------------------------
__global__ void k_zero(float4* aggr, int na4, float* stats, int ns) {
  int stride = gridDim.x * blockDim.x;
  float4 z = {0.f, 0.f, 0.f, 0.f};
  for (int i = blockIdx.x * blockDim.x + threadIdx.x; i < na4; i += stride) aggr[i] = z;
  for (int i = blockIdx.x * blockDim.x + threadIdx.x; i < ns; i += stride) stats[i] = 0.f;
}

// ---------------------------------------------------------------------------
// K1: fused node GEMMs  Ax,Bx,Dx,Ex = x @ {WA,WB,WD,WE}^T + bias
// One 16-row tile per block; 4 waves, one weight matrix each.
// ---------------------------------------------------------------------------
__global__ void k_node_gemm4(const float* __restrict__ x,
                             const float* __restrict__ WA, const float* __restrict__ bA,
                             const float* __restrict__ WB, const float* __restrict__ bB,
                             const float* __restrict__ WD, const float* __restrict__ bD,
                             const float* __restrict__ WE, const float* __restrict__ bE,
                             float* __restrict__ Ax, float* __restrict__ Bx,
                             float* __restrict__ Dx, float* __restrict__ Ex, int N) {
  int wave = threadIdx.x >> 5;
  int lane = threadIdx.x & 31;
  int row0 = blockIdx.x * 16;
  if (row0 >= N) return;
  bool full = (row0 + 16 <= N);

  const float* W; const float* bias; float* out;
  switch (wave) {
    case 0: W = WA; bias = bA; out = Ax; break;
    case 1: W = WB; bias = bB; out = Bx; break;
    case 2: W = WD; bias = bD; out = Dx; break;
    default:W = WE; bias = bE; out = Ex; break;
  }

  for (int n0 = 0; n0 < 64; n0 += 16) {
    v8f acc = {};
#pragma unroll
    for (int k0 = 0; k0 < 64; k0 += 4) {
      v2f a = load_a(x, row0, k0, lane, N, 64);
      v2f b = load_bt(W, n0, k0, lane, 64);
      acc = wmma4(a, b, acc);
    }
    float bv = bias[n0 + (lane & 15)];
#pragma unroll
    for (int r = 0; r < 8; ++r) acc[r] += bv;
    if (full) store_d_full(out, row0, n0, lane, acc, 64);
    else      store_d_guard(out, row0, n0, lane, acc, N, 64);
  }
}

// ---------------------------------------------------------------------------
// K2: edge kernel. One 16-edge tile per wave.
//   Ce = edge_attr @ WC^T + bC               (WMMA)
//   e_ij = Dx[row] + Ex[col] + Ce            -> written into e-half of d_out
//   msg  = sigmoid(e_ij) * Bx[col]           -> atomic scatter-add into aggr
//   column sum/sumsq of e_ij                 -> stats_e (for BN over E rows)
// ---------------------------------------------------------------------------
__global__ void k_edge(const float* __restrict__ ea,
                       const int* __restrict__ row_idx, const int* __restrict__ col_idx,
                       const float* __restrict__ WC, const float* __restrict__ bC,
                       const float* __restrict__ Dx, const float* __restrict__ Ex,
                       const float* __restrict__ Bx,
                       float* __restrict__ e_out, float* __restrict__ aggr,
                       float* __restrict__ stats_e, int E) {
  int wid  = blockIdx.x * (blockDim.x >> 5) + (threadIdx.x >> 5);
  int lane = threadIdx.x & 31;
  int e0   = wid * 16;
  if (e0 >= E) return;
  bool full = (e0 + 16 <= E);

  // indices for the 8 edges this half-wave's accumulator rows map onto
  int base = e0 + (lane >> 4) * 8;
  int rows[8], cols[8];
#pragma unroll
  for (int r = 0; r < 8; ++r) {
    int e = base + r; if (e > E - 1) e = E - 1;
    rows[r] = row_idx[e];
    cols[r] = col_idx[e];
  }

  for (int t = 0; t < 4; ++t) {
    int n0 = t * 16;
    v8f acc = {};
#pragma unroll
    for (int k0 = 0; k0 < 64; k0 += 4) {
      v2f a = load_a(ea, e0, k0, lane, E, 64);
      v2f b = load_bt(WC, n0, k0, lane, 64);
      acc = wmma4(a, b, acc);
    }
    int   feat = n0 + (lane & 15);
    float bcv  = bC[feat];
    float esum = 0.f, esq = 0.f;
    if (full) {
#pragma unroll
      for (int r = 0; r < 8; ++r) {
        int e = base + r;
        float eij = acc[r] + bcv + Dx[(size_t)rows[r] * 64 + feat]
                                 + Ex[(size_t)cols[r] * 64 + feat];
        e_out[(size_t)e * 64 + feat] = eij;
        esum += eij; esq += eij * eij;
        float gate = 1.f / (1.f + __expf(-eij));
        atomicAdd(&aggr[(size_t)rows[r] * 64 + feat],
                  gate * Bx[(size_t)cols[r] * 64 + feat]);
      }
    } else {
#pragma unroll
      for (int r = 0; r < 8; ++r) {
        int e = base + r;
        float eij = acc[r] + bcv + Dx[(size_t)rows[r] * 64 + feat]
                                 + Ex[(size_t)cols[r] * 64 + feat];
        if (e < E) {
          e_out[(size_t)e * 64 + feat] = eij;
          esum += eij; esq += eij * eij;
          float gate = 1.f / (1.f + __expf(-eij));
          atomicAdd(&aggr[(size_t)rows[r] * 64 + feat],
                    gate * Bx[(size_t)cols[r] * 64 + feat]);
        }
      }
    }
    atomicAdd(&stats_e[feat],      esum);
    atomicAdd(&stats_e[64 + feat], esq);
  }
}

// ---------------------------------------------------------------------------
// K3a: x_t = Ax + aggr (+ column stats), float4 path
// ---------------------------------------------------------------------------
__global__ void k_xt(const float4* __restrict__ Ax, const float4* __restrict__ aggr,
                     float4* __restrict__ xt, float* __restrict__ stats, int total4) {
  __shared__ float ssum[64], ssq[64];
  if (threadIdx.x < 64) { ssum[threadIdx.x] = 0.f; ssq[threadIdx.x] = 0.f; }
  __syncthreads();
  for (int i = blockIdx.x * blockDim.x + threadIdx.x; i < total4;
       i += gridDim.x * blockDim.x) {
    float4 a = Ax[i], g = aggr[i];
    float4 v = {a.x + g.x, a.y + g.y, a.z + g.z, a.w + g.w};
    xt[i] = v;
    int f0 = (i & 15) * 4;
    atomicAdd(&ssum[f0 + 0], v.x); atomicAdd(&ssq[f0 + 0], v.x * v.x);
    atomicAdd(&ssum[f0 + 1], v.y); atomicAdd(&ssq[f0 + 1], v.y * v.y);
    atomicAdd(&ssum[f0 + 2], v.z); atomicAdd(&ssq[f0 + 2], v.z * v.z);
    atomicAdd(&ssum[f0 + 3], v.w); atomicAdd(&ssq[f0 + 3], v.w * v.w);
  }
  __syncthreads();
  if (threadIdx.x < 64) {
    atomicAdd(&stats[threadIdx.x],      ssum[threadIdx.x]);
    atomicAdd(&stats[64 + threadIdx.x], ssq[threadIdx.x]);
  }
}

// ---------------------------------------------------------------------------
// K3b: x_final = x + relu(BN(x_t))  (+ column stats for n1), float4 path
// ---------------------------------------------------------------------------
__global__ void k_xfinal(const float4* __restrict__ x, const float4* __restrict__ xt,
                         const float* __restrict__ st_xt,
                         const float* __restrict__ g, const float* __restrict__ b,
                         float4* __restrict__ xf, float* __restrict__ st_n1,
                         int total4, float invRows) {
  __shared__ float ssum[64], ssq[64];
  if (threadIdx.x < 64) { ssum[threadIdx.x] = 0.f; ssq[threadIdx.x] = 0.f; }
  __syncthreads();
  for (int i = blockIdx.x * blockDim.x + threadIdx.x; i < total4;
       i += gridDim.x * blockDim.x) {
    int f0 = (i & 15) * 4;
    float4 t = xt[i], xv = x[i], o;
    float* tp = &t.x; const float* xp = &xv.x; float* op = &o.x;
#pragma unroll
    for (int c = 0; c < 4; ++c) {
      int f = f0 + c;
      float mean = st_xt[f] * invRows;
      float var  = st_xt[64 + f] * invRows - mean * mean;
      float xn   = g[f] * (tp[c] - mean) * rsqrtf(var + EPS) + b[f];
      float v    = xp[c] + fmaxf(xn, 0.f);
      op[c] = v;
      atomicAdd(&ssum[f], v); atomicAdd(&ssq[f], v * v);
    }
    xf[i] = o;
  }
  __syncthreads();
  if (threadIdx.x < 64) {
    atomicAdd(&st_n1[threadIdx.x],      ssum[threadIdx.x]);
    atomicAdd(&st_n1[64 + threadIdx.x], ssq[threadIdx.x]);
  }
}

// ---------------------------------------------------------------------------
// K4: FFN  y = x_final + ( relu(BN1(x_final) @ W1^T + b1) @ W2^T + b2 )
// One 16-row tile per wave; 4 waves per block; 16x128 intermediate staged in
// wave-private LDS to re-fragment for the K=128 second GEMM. (+ stats for n2)
// ---------------------------------------------------------------------------
__global__ void k_ffn(const float* __restrict__ xf, const float* __restrict__ st_n1,
                      const float* __restrict__ g1, const float* __restrict__ b1n,
                      const float* __restrict__ W1, const float* __restrict__ b1,
                      const float* __restrict__ W2, const float* __restrict__ b2,
                      float* __restrict__ y, float* __restrict__ st_n2,
                      int N, float invN) {
  __shared__ float h1s[4][16 * 128];
  int wip  = threadIdx.x >> 5;
  int lane = threadIdx.x & 31;
  int tile = blockIdx.x * 4 + wip;
  int row0 = tile * 16;
  if (row0 >= N) return;
  bool full = (row0 + 16 <= N);

  int m  = lane & 15;
  int kk = (lane >> 4) << 1;

  // ---- GEMM1: BN1(xf) @ W1^T + b1, relu -> LDS (16 x 128) ----
  for (int n0 = 0; n0 < 128; n0 += 16) {
    v8f acc = {};
#pragma unroll
    for (int k0 = 0; k0 < 64; k0 += 4) {
      int r  = row0 + m; if (r > N - 1) r = N - 1;
      int kA = k0 + kk;
      float m0 = st_n1[kA]      * invN, m1 = st_n1[kA + 1]      * invN;
      float v0 = st_n1[64+kA]   * invN - m0 * m0;
      float v1 = st_n1[64+kA+1] * invN - m1 * m1;
      float x0 = xf[(size_t)r * 64 + kA];
      float x1 = xf[(size_t)r * 64 + kA + 1];
      v2f a;
      a.x = g1[kA]     * (x0 - m0) * rsqrtf(v0 + EPS) + b1n[kA];
      a.y = g1[kA + 1] * (x1 - m1) * rsqrtf(v1 + EPS) + b1n[kA + 1];
      v2f b = load_bt(W1, n0, k0, lane, 64);
      acc = wmma4(a, b, acc);
    }
    float bv   = b1[n0 + m];
    int   half = lane >> 4;
#pragma unroll
    for (int r = 0; r < 8; ++r)
      h1s[wip][(r + half * 8) * 128 + n0 + m] = fmaxf(acc[r] + bv, 0.f);
  }

  // ---- GEMM2: h1 @ W2^T + b2, residual, stats ----
  for (int n0 = 0; n0 < 64; n0 += 16) {
    v8f acc = {};
#pragma unroll
    for (int k0 = 0; k0 < 128; k0 += 4) {
      v2f a;
      a.x = h1s[wip][m * 128 + k0 + kk];
      a.y = h1s[wip][m * 128 + k0 + kk + 1];
      v2f b = load_bt(W2, n0, k0, lane, 128);
      acc = wmma4(a, b, acc);
    }
    int   feat = n0 + m;
    float bv   = b2[feat];
    int   half = lane >> 4;
    float lsum = 0.f, lsq = 0.f;
    if (full) {
#pragma unroll
      for (int r = 0; r < 8; ++r) {
        int row = row0 + r + half * 8;
        float yv = xf[(size_t)row * 64 + feat] + acc[r] + bv;
        y[(size_t)row * 64 + feat] = yv;
        lsum += yv; lsq += yv * yv;
      }
    } else {
#pragma unroll
      for (int r = 0; r < 8; ++r) {
        int row = row0 + r + half * 8;
        if (row < N) {
          float yv = xf[(size_t)row * 64 + feat] + acc[r] + bv;
          y[(size_t)row * 64 + feat] = yv;
          lsum += yv; lsq += yv * yv;
        }
      }
    }
    atomicAdd(&st_n2[feat],      lsum);
    atomicAdd(&st_n2[64 + feat], lsq);
  }
}

// ---------------------------------------------------------------------------
// K5: x_out = BN2(y), float4 path
// ---------------------------------------------------------------------------
__global__ void k_bn_out(const float4* __restrict__ y, const float* __restrict__ st,
                         const float* __restrict__ g, const float* __restrict__ b,
                         float4* __restrict__ out, int total4, float invRows) {
  for (int i = blockIdx.x * blockDim.x + threadIdx.x; i < total4;
       i += gridDim.x * blockDim.x) {
    int f0 = (i & 15) * 4;
    float4 v = y[i], o;
    const float* vp = &v.x; float* op = &o.x;
#pragma unroll
    for (int c = 0; c < 4; ++c) {
      int f = f0 + c;
      float mean = st[f] * invRows;
      float var  = st[64 + f] * invRows - mean * mean;
      op[c] = g[f] * (vp[c] - mean) * rsqrtf(var + EPS) + b[f];
    }
    out[i] = o;
  }
}

// ---------------------------------------------------------------------------
// K6: e_final = edge_attr + relu(BN_e(e_ij))  (in place, e-half of d_out),
// float4 path: 410 MB of the total traffic goes through this kernel.
// ---------------------------------------------------------------------------
__global__ void k_efinal(const float4* __restrict__ ea, const float* __restrict__ st,
                         const float* __restrict__ g, const float* __restrict__ b,
                         float4* __restrict__ e_io, int total4, float invRows) {
  for (int i = blockIdx.x * blockDim.x + threadIdx.x; i < total4;
       i += gridDim.x * blockDim.x) {
    int f0 = (i & 15) * 4;
    float4 v = e_io[i], av = ea[i], o;
    const float* vp = &v.x; const float* ap = &av.x; float* op = &o.x;
#pragma unroll
    for (int c = 0; c < 4; ++c) {
      int f = f0 + c;
      float mean = st[f] * invRows;
      float var  = st[64 + f] * invRows - mean * mean;
      float en   = g[f] * (vp[c] - mean) * rsqrtf(var + EPS) + b[f];
      op[c] = ap[c] + fmaxf(en, 0.f);
    }
    e_io[i] = o;
  }
}

// ---------------------------------------------------------------------------
extern "C" void kernel_launch(void* const* d_in, const int* in_sizes, int n_in,
                              void* d_out, int out_size, void* d_ws, size_t ws_size,
                              hipStream_t stream) {
  const float* x   = (const float*)d_in[0];
  const int*   ei  = (const int*)  d_in[1];
  const float* ea  = (const float*)d_in[2];
  const float* WA  = (const float*)d_in[3];  const float* bA  = (const float*)d_in[4];
  const float* WB  = (const float*)d_in[5];  const float* bB  = (const float*)d_in[6];
  const float* WC  = (const float*)d_in[7];  const float* bC  = (const float*)d_in[8];
  const float* WD  = (const float*)d_in[9];  const float* bD  = (const float*)d_in[10];
  const float* WE  = (const float*)d_in[11]; const float* bE  = (const float*)d_in[12];
  const float* g_x = (const float*)d_in[13]; const float* b_x = (const float*)d_in[14];
  const float* g_e = (const float*)d_in[15]; const float* b_e = (const float*)d_in[16];
  const float* g_n1= (const float*)d_in[17]; const float* b_n1= (const float*)d_in[18];
  const float* W1  = (const float*)d_in[19]; const float* b1  = (const float*)d_in[20];
  const float* W2  = (const float*)d_in[21]; const float* b2  = (const float*)d_in[22];
  const float* g_n2= (const float*)d_in[23]; const float* b_n2= (const float*)d_in[24];

  const int N = in_sizes[0] / 64;
  const int E = in_sizes[2] / 64;
  const int* row = ei;
  const int* col = ei + E;

  float* ws  = (float*)d_ws;
  size_t nd  = (size_t)N * 64;
  float* Ax   = ws + 0 * nd;
  float* Bx   = ws + 1 * nd;
  float* Dx   = ws + 2 * nd;
  float* Ex   = ws + 3 * nd;
  float* aggr = ws + 4 * nd;
  float* xt   = ws + 5 * nd;
  float* xf   = ws + 6 * nd;
  float* y    = ws + 7 * nd;
  float* st_xt = ws + 8 * nd;        // 128
  float* st_n1 = st_xt + 128;        // 128
  float* st_n2 = st_n1 + 128;        // 128
  float* st_e  = st_n2 + 128;        // 128

  float* out_x = (float*)d_out;
  float* out_e = out_x + nd;

  const int ntot4 = (int)(nd / 4);
  const int etot4 = (E * 64) / 4;
  const int ntiles = (N + 15) / 16;
  const int etiles = (E + 15) / 16;

  k_zero<<<1024, 256, 0, stream>>>((float4*)aggr, ntot4, st_xt, 512);

  k_node_gemm4<<<ntiles, 128, 0, stream>>>(x, WA, bA, WB, bB, WD, bD, WE, bE,
                                           Ax, Bx, Dx, Ex, N);

  k_edge<<<(etiles + 7) / 8, 256, 0, stream>>>(ea, row, col, WC, bC, Dx, Ex, Bx,
                                               out_e, aggr, st_e, E);

  k_xt<<<1024, 256, 0, stream>>>((const float4*)Ax, (const float4*)aggr,
                                 (float4*)xt, st_xt, ntot4);

  k_xfinal<<<1024, 256, 0, stream>>>((const float4*)x, (const float4*)xt, st_xt,
                                     g_x, b_x, (float4*)xf, st_n1,
                                     ntot4, 1.0f / (float)N);

  k_ffn<<<(ntiles + 3) / 4, 128, 0, stream>>>(xf, st_n1, g_n1, b_n1, W1, b1,
                                              W2, b2, y, st_n2, N, 1.0f / (float)N);

  k_bn_out<<<1024, 256, 0, stream>>>((const float4*)y, st_n2, g_n2, b_n2,
                                     (float4*)out_x, ntot4, 1.0f / (float)N);

  k_efinal<<<4096, 256, 0, stream>>>((const float4*)ea, st_e, g_e, b_e,
                                     (float4*)out_e, etot4, 1.0f / (float)E);
}